// GNN2_63015760167405
// MI455X (gfx1250) — compile-verified
//
#include <hip/hip_runtime.h>
#include <math.h>

#define HID 64
#define NEG_SLOPE 0.2f
#define N_GRAPHS 64

typedef float v2f __attribute__((ext_vector_type(2)));
typedef float v8f __attribute__((ext_vector_type(8)));

// ---- order-preserving float<->uint encoding for atomicMax on f32 ----
__device__ __forceinline__ unsigned enc_f32(float f) {
    unsigned u = __float_as_uint(f);
    return (u & 0x80000000u) ? ~u : (u | 0x80000000u);
}
__device__ __forceinline__ float dec_f32(unsigned u) {
    unsigned b = (u & 0x80000000u) ? (u ^ 0x80000000u) : ~u;
    return __uint_as_float(b);
}

// ---------------------------------------------------------------------
// K1: h = x @ W using V_WMMA_F32_16X16X4_F32. One wave = 16x64 tile.
// ---------------------------------------------------------------------
__global__ __launch_bounds__(128) void k_gemm_wmma(const float* __restrict__ x,
                                                   const float* __restrict__ W,
                                                   float* __restrict__ h, int N) {
    const int wave  = threadIdx.x >> 5;
    const int lane  = threadIdx.x & 31;
    const int rowBase = (blockIdx.x * 4 + wave) * 16;
    const int mrow  = lane & 15;
    const int khalf = (lane >> 4) << 1;   // 0 or 2
    const int ncol  = lane & 15;

    int arow = rowBase + mrow;
    if (arow >= N) arow = N - 1;          // clamp (keeps EXEC all-ones)
    const float* xr = x + (size_t)arow * HID;

    v8f acc0 = {}, acc1 = {}, acc2 = {}, acc3 = {};

#pragma unroll
    for (int kk = 0; kk < 16; ++kk) {
        const int k0 = kk * 4 + khalf;
        v2f a;
        a.x = xr[k0];
        a.y = xr[k0 + 1];
        const float* w0 = W + (size_t)k0 * HID;
        const float* w1 = W + (size_t)(k0 + 1) * HID;
        v2f b;
        b.x = w0[ncol +  0]; b.y = w1[ncol +  0];
        acc0 = __builtin_amdgcn_wmma_f32_16x16x4_f32(false, a, false, b, (short)0, acc0, false, false);
        b.x = w0[ncol + 16]; b.y = w1[ncol + 16];
        acc1 = __builtin_amdgcn_wmma_f32_16x16x4_f32(false, a, false, b, (short)0, acc1, false, false);
        b.x = w0[ncol + 32]; b.y = w1[ncol + 32];
        acc2 = __builtin_amdgcn_wmma_f32_16x16x4_f32(false, a, false, b, (short)0, acc2, false, false);
        b.x = w0[ncol + 48]; b.y = w1[ncol + 48];
        acc3 = __builtin_amdgcn_wmma_f32_16x16x4_f32(false, a, false, b, (short)0, acc3, false, false);
    }

    const int mbase = (lane >> 4) << 3;   // 0 or 8
#pragma unroll
    for (int i = 0; i < 8; ++i) {
        int row = rowBase + mbase + i;
        if (row < N) {
            float* hr = h + (size_t)row * HID + ncol;
            hr[0]  = acc0[i];
            hr[16] = acc1[i];
            hr[32] = acc2[i];
            hr[48] = acc3[i];
        }
    }
}

// ---------------------------------------------------------------------
// K2: a_s[i] = h[i]·att_src ; a_d[i] = h[i]·att_dst (one wave per node)
// ---------------------------------------------------------------------
__global__ __launch_bounds__(256) void k_att(const float* __restrict__ h,
                                             const float* __restrict__ att_src,
                                             const float* __restrict__ att_dst,
                                             float* __restrict__ a_s,
                                             float* __restrict__ a_d, int N) {
    int wid  = (int)((blockIdx.x * blockDim.x + threadIdx.x) >> 5);
    int lane = threadIdx.x & 31;
    if (wid >= N) return;                 // uniform per wave
    int c = lane * 2;
    const float* hr = h + (size_t)wid * HID + c;
    float h0 = hr[0], h1 = hr[1];
    float s = h0 * att_src[c] + h1 * att_src[c + 1];
    float d = h0 * att_dst[c] + h1 * att_dst[c + 1];
#pragma unroll
    for (int m = 16; m >= 1; m >>= 1) {
        s += __shfl_xor(s, m, 32);
        d += __shfl_xor(d, m, 32);
    }
    if (lane == 0) { a_s[wid] = s; a_d[wid] = d; }
}

// ---------------------------------------------------------------------
// K3: zero scratch
// ---------------------------------------------------------------------
__global__ void k_zero(float* __restrict__ p, long long n) {
    long long i = (long long)blockIdx.x * blockDim.x + threadIdx.x;
    if (i < n) p[i] = 0.0f;
}

// ---------------------------------------------------------------------
// K4: edge pass 1 — logits + segment max (uint-encoded atomicMax)
// ---------------------------------------------------------------------
__global__ __launch_bounds__(256) void k_edge_max(const int* __restrict__ ei, int E, int N,
                                                  const float* __restrict__ a_s,
                                                  const float* __restrict__ a_d,
                                                  float* __restrict__ e_buf,
                                                  unsigned* __restrict__ m_enc) {
    int i = blockIdx.x * blockDim.x + threadIdx.x;
    int EP = E + N;
    if (i >= EP) return;
    int src, dst;
    if (i < E) { src = ei[i]; dst = ei[E + i]; }
    else       { src = dst = i - E; }                 // self loop
    float e = a_s[src] + a_d[dst];
    e = e > 0.0f ? e : NEG_SLOPE * e;                 // leaky relu
    e_buf[i] = e;
    atomicMax(m_enc + dst, enc_f32(e));
}

// ---------------------------------------------------------------------
// K5: edge pass 2 — ex=exp(e-m); denom[dst]+=ex; hacc[dst]+=ex*h[src]
//     (softmax denominator division deferred to node pass)
//     16 lanes per edge, 4 channels each (float4 gather of h[src]).
// ---------------------------------------------------------------------
__global__ __launch_bounds__(256) void k_edge_acc(const int* __restrict__ ei, int E, int N,
                                                  const float* __restrict__ e_buf,
                                                  const unsigned* __restrict__ m_enc,
                                                  const float* __restrict__ h,
                                                  float* __restrict__ denom,
                                                  float* __restrict__ hacc) {
    int t    = blockIdx.x * blockDim.x + threadIdx.x;
    int edge = t >> 4;
    int lane = t & 15;
    int EP   = E + N;
    if (edge >= EP) return;
    int src, dst;
    if (edge < E) { src = ei[edge]; dst = ei[E + edge]; }
    else          { src = dst = edge - E; }
    float ex = expf(e_buf[edge] - dec_f32(m_enc[dst]));
    if (lane == 0) atomicAdd(denom + dst, ex);
    const float4 hv = *(const float4*)(h + (size_t)src * HID + lane * 4);
    float* ha = hacc + (size_t)dst * HID + lane * 4;
    atomicAdd(ha + 0, ex * hv.x);
    atomicAdd(ha + 1, ex * hv.y);
    atomicAdd(ha + 2, ex * hv.z);
    atomicAdd(ha + 3, ex * hv.w);
}

// ---------------------------------------------------------------------
// K6: node finalize + mean-pool accumulation.
//     64 nodes/block, 4 threads/node x 16 channels. batch is sorted, so
//     accumulate in LDS and flush only the graphs this block touches.
// ---------------------------------------------------------------------
__global__ __launch_bounds__(256) void k_node_pool(const float* __restrict__ hacc,
                                                   const float* __restrict__ denom,
                                                   const float* __restrict__ bias,
                                                   const int* __restrict__ batch,
                                                   int N,
                                                   float* __restrict__ pool,
                                                   float* __restrict__ cnt) {
    __shared__ float lp[N_GRAPHS * HID];
    __shared__ float lc[N_GRAPHS];
    __shared__ int   gr[2];
    for (int i = threadIdx.x; i < N_GRAPHS * HID; i += 256) lp[i] = 0.0f;
    if (threadIdx.x < N_GRAPHS) lc[threadIdx.x] = 0.0f;
    int nodeBase = blockIdx.x * 64;
    if (threadIdx.x == 0) {
        int f = nodeBase < N ? nodeBase : N - 1;
        int l = nodeBase + 63; if (l >= N) l = N - 1;
        gr[0] = batch[f];
        gr[1] = batch[l];
    }
    __syncthreads();

    int node = nodeBase + (threadIdx.x >> 2);
    int part = threadIdx.x & 3;
    if (node < N) {
        int g = batch[node];
        float inv = 1.0f / denom[node];
        const float* ha = hacc + (size_t)node * HID;
#pragma unroll
        for (int j = 0; j < 16; ++j) {
            int c = part * 16 + j;
            float v = ha[c] * inv + bias[c];
            v = v > 0.0f ? v : 0.0f;                  // relu
            atomicAdd(&lp[g * HID + c], v);           // ds_add_f32
        }
        if (part == 0) atomicAdd(&lc[g], 1.0f);
    }
    __syncthreads();

    int g0 = gr[0], g1 = gr[1];
    int total = (g1 - g0 + 1) * HID;
    for (int i = threadIdx.x; i < total; i += 256) {
        int g = g0 + i / HID;
        int c = i % HID;
        float v = lp[g * HID + c];
        if (v != 0.0f) atomicAdd(pool + g * HID + c, v);
    }
    if (threadIdx.x <= (unsigned)(g1 - g0)) {
        float v = lc[g0 + threadIdx.x];
        if (v != 0.0f) atomicAdd(cnt + g0 + threadIdx.x, v);
    }
}

// ---------------------------------------------------------------------
// K7: z1 = relu((pool @ w1)/cnt + b1)   (pooled mean folded into /cnt)
// ---------------------------------------------------------------------
__global__ __launch_bounds__(256) void k_mlp1(const float* __restrict__ pool,
                                              const float* __restrict__ cnt,
                                              const float* __restrict__ w1,
                                              const float* __restrict__ b1,
                                              float* __restrict__ z1) {
    int idx = blockIdx.x * blockDim.x + threadIdx.x;
    if (idx >= N_GRAPHS * HID) return;
    int g = idx >> 6, c = idx & 63;
    float cn = cnt[g]; cn = cn > 1.0f ? cn : 1.0f;
    const float* pr = pool + g * HID;
    float s = 0.0f;
#pragma unroll 8
    for (int k = 0; k < HID; ++k) s += pr[k] * w1[k * HID + c];
    s = s / cn + b1[c];
    z1[idx] = s > 0.0f ? s : 0.0f;
}

// ---------------------------------------------------------------------
// K8: out = sigmoid(z1 @ w2 + b2)
// ---------------------------------------------------------------------
__global__ void k_mlp2(const float* __restrict__ z1,
                       const float* __restrict__ w2,
                       const float* __restrict__ b2,
                       float* __restrict__ out) {
    int g = threadIdx.x;
    if (g >= N_GRAPHS) return;
    float s = b2[0];
    const float* zr = z1 + (size_t)g * HID;
#pragma unroll 8
    for (int k = 0; k < HID; ++k) s += zr[k] * w2[k];
    out[g] = 1.0f / (1.0f + expf(-s));
}

// ---------------------------------------------------------------------
extern "C" void kernel_launch(void* const* d_in, const int* in_sizes, int n_in,
                              void* d_out, int out_size, void* d_ws, size_t ws_size,
                              hipStream_t stream) {
    const float* x        = (const float*)d_in[0];
    const int*   ei       = (const int*)  d_in[1];
    const int*   batch    = (const int*)  d_in[2];
    const float* W        = (const float*)d_in[3];
    const float* att_src  = (const float*)d_in[4];
    const float* att_dst  = (const float*)d_in[5];
    const float* bias_gat = (const float*)d_in[6];
    const float* w1       = (const float*)d_in[7];
    const float* b1       = (const float*)d_in[8];
    const float* w2       = (const float*)d_in[9];
    const float* b2       = (const float*)d_in[10];
    float* out = (float*)d_out;

    const int N  = in_sizes[0] / HID;
    const int E  = in_sizes[1] / 2;
    const int EP = E + N;

    // workspace layout (floats)
    float* ws = (float*)d_ws;
    size_t off = 0;
    float*    h     = ws + off; off += (size_t)N * HID;
    float*    a_s   = ws + off; off += N;
    float*    a_d   = ws + off; off += N;
    float*    e_buf = ws + off; off += EP;
    unsigned* m_enc = (unsigned*)(ws + off); off += N;   // start of zeroed region
    float*    denom = ws + off; off += N;
    float*    hacc  = ws + off; off += (size_t)N * HID;
    float*    pool  = ws + off; off += N_GRAPHS * HID;
    float*    cnt   = ws + off; off += N_GRAPHS;
    float*    z1    = ws + off; off += N_GRAPHS * HID;

    // zero the contiguous region m_enc..z1 (0.0f bits == 0u for m_enc)
    long long zn = (long long)((ws + off) - (float*)m_enc);
    k_zero<<<(unsigned)((zn + 255) / 256), 256, 0, stream>>>((float*)m_enc, zn);

    // h = x @ W  (WMMA fp32)
    k_gemm_wmma<<<(N + 63) / 64, 128, 0, stream>>>(x, W, h, N);

    // attention logit components
    k_att<<<(N + 7) / 8, 256, 0, stream>>>(h, att_src, att_dst, a_s, a_d, N);

    // edge pass 1: logits + segment max
    k_edge_max<<<(EP + 255) / 256, 256, 0, stream>>>(ei, E, N, a_s, a_d, e_buf, m_enc);

    // edge pass 2: softmax numerators + fused weighted scatter
    long long t5 = (long long)EP * 16;
    k_edge_acc<<<(unsigned)((t5 + 255) / 256), 256, 0, stream>>>(ei, E, N, e_buf, m_enc,
                                                                 h, denom, hacc);

    // node finalize + mean pool
    k_node_pool<<<(N + 63) / 64, 256, 0, stream>>>(hacc, denom, bias_gat, batch, N, pool, cnt);

    // MLP head
    k_mlp1<<<(N_GRAPHS * HID + 255) / 256, 256, 0, stream>>>(pool, cnt, w1, b1, z1);
    k_mlp2<<<1, 64, 0, stream>>>(z1, w2, b2, out);
}